// GCN_66383014527488
// MI455X (gfx1250) — compile-verified
//
#include <hip/hip_runtime.h>
#include <hip/hip_bf16.h>

typedef __attribute__((ext_vector_type(2))) float v2f;
typedef __attribute__((ext_vector_type(8))) float v8f;

#define NNODES 50000
#define NEDGES 800000
#define FDIM   256             // K and Nout for layers 0/1 (power of two -> shifts)
#define MTILES (NNODES / 16)   // 3125, exact
#define MPAIRS ((MTILES + 1) / 2)   // 1563 (last pair duplicates tile 3124)

// ---------------- degree / normalization ----------------

__global__ __launch_bounds__(256) void deg_init_kernel(float* __restrict__ deg) {
    int i = blockIdx.x * blockDim.x + threadIdx.x;
    if (i < NNODES) deg[i] = 1.0f;   // self-loop contributes 1
}

__global__ __launch_bounds__(256) void deg_count_kernel(const int* __restrict__ ei,
                                                        float* __restrict__ deg) {
    int i = blockIdx.x * blockDim.x + threadIdx.x;
    if (i < NEDGES) atomicAdd(&deg[ei[NEDGES + i]], 1.0f);   // dst row
}

__global__ __launch_bounds__(256) void dinv_kernel(float* __restrict__ deg) {
    int i = blockIdx.x * blockDim.x + threadIdx.x;
    if (i < NNODES) {
        float d = deg[i];
        deg[i] = (d > 0.0f) ? __frsqrt_rn(d) : 0.0f;    // in place: deg -> dinv
    }
}

// ---------------- GEMM via V_WMMA_F32_16X16X4_F32 ----------------
// C[N x 256] = act(A)[N x 256] @ W[256 x 256], row-major.
// One wave computes a 32x64 block of C (2 M-tiles x 4 N-tiles, 8 accumulators).
// Each B fragment feeds 2 wmma; each A fragment feeds 4 wmma.

template <int RELU>
__global__ __launch_bounds__(256) void gemm_wmma_f32_kernel(const float* __restrict__ A,
                                                            const float* __restrict__ W,
                                                            float* __restrict__ C) {
    const int wid  = (int)((blockIdx.x * 256u + threadIdx.x) >> 5);
    if (wid >= MPAIRS * 4) return;
    const int lane = threadIdx.x & 31;
    const int mp   = wid >> 2;                 // M-tile pair
    const int tg   = wid & 3;                  // 64-column group

    const int tm0 = (2 * mp     < MTILES) ? 2 * mp     : MTILES - 1;
    const int tm1 = (2 * mp + 1 < MTILES) ? 2 * mp + 1 : MTILES - 1;

    const int l15   = lane & 15;
    const int khalf = (lane >> 4) << 1;        // lanes 16-31 -> k+2,k+3
    const int ncol0 = (tg << 6) + l15;         // first of 4 columns this lane touches

    const float* __restrict__ arow0 = A + (((unsigned)(tm0 * 16 + l15)) << 8);
    const float* __restrict__ arow1 = A + (((unsigned)(tm1 * 16 + l15)) << 8);

    v8f acc00 = {}, acc01 = {}, acc02 = {}, acc03 = {};
    v8f acc10 = {}, acc11 = {}, acc12 = {}, acc13 = {};

#pragma unroll 2
    for (int k = 0; k < FDIM; k += 4) {
        v2f a0 = *reinterpret_cast<const v2f*>(arow0 + k + khalf);
        v2f a1 = *reinterpret_cast<const v2f*>(arow1 + k + khalf);
        if (RELU) {
            a0.x = fmaxf(a0.x, 0.0f); a0.y = fmaxf(a0.y, 0.0f);
            a1.x = fmaxf(a1.x, 0.0f); a1.y = fmaxf(a1.y, 0.0f);
        }

        const float* __restrict__ w0 = W + (((unsigned)(k + khalf)) << 8) + ncol0;
        const float* __restrict__ w1 = w0 + FDIM;   // row k+khalf+1
        v2f b0; b0.x = w0[0];  b0.y = w1[0];
        v2f b1; b1.x = w0[16]; b1.y = w1[16];
        v2f b2; b2.x = w0[32]; b2.y = w1[32];
        v2f b3; b3.x = w0[48]; b3.y = w1[48];

        acc00 = __builtin_amdgcn_wmma_f32_16x16x4_f32(false, a0, false, b0, (short)0, acc00, false, false);
        acc10 = __builtin_amdgcn_wmma_f32_16x16x4_f32(false, a1, false, b0, (short)0, acc10, false, false);
        acc01 = __builtin_amdgcn_wmma_f32_16x16x4_f32(false, a0, false, b1, (short)0, acc01, false, false);
        acc11 = __builtin_amdgcn_wmma_f32_16x16x4_f32(false, a1, false, b1, (short)0, acc11, false, false);
        acc02 = __builtin_amdgcn_wmma_f32_16x16x4_f32(false, a0, false, b2, (short)0, acc02, false, false);
        acc12 = __builtin_amdgcn_wmma_f32_16x16x4_f32(false, a1, false, b2, (short)0, acc12, false, false);
        acc03 = __builtin_amdgcn_wmma_f32_16x16x4_f32(false, a0, false, b3, (short)0, acc03, false, false);
        acc13 = __builtin_amdgcn_wmma_f32_16x16x4_f32(false, a1, false, b3, (short)0, acc13, false, false);
    }

    // C/D layout: VGPR r -> M = tm*16 + r + (lane>=16 ? 8 : 0), N = column
    const unsigned hi8 = (unsigned)((lane >> 4) << 3);
    float* __restrict__ cp0 = C + (((unsigned)(tm0 * 16) + hi8) << 8) + ncol0;
    float* __restrict__ cp1 = C + (((unsigned)(tm1 * 16) + hi8) << 8) + ncol0;
#pragma unroll
    for (int r = 0; r < 8; ++r) {
        float* row0 = cp0 + ((unsigned)r << 8);
        row0[0]  = acc00[r];
        row0[16] = acc01[r];
        row0[32] = acc02[r];
        row0[48] = acc03[r];
        float* row1 = cp1 + ((unsigned)r << 8);
        row1[0]  = acc10[r];
        row1[16] = acc11[r];
        row1[32] = acc12[r];
        row1[48] = acc13[r];
    }
}

// ---------------- aggregation ----------------
// agg[i][f] = dinv[i]^2 * t[i][f] + b[f]   (self-loop + bias), float4 per thread
__global__ __launch_bounds__(256) void agg_init_kernel(const float* __restrict__ t,
                                                       const float* __restrict__ dinv,
                                                       const float* __restrict__ bias,
                                                       float* __restrict__ agg) {
    unsigned gid = blockIdx.x * 256u + threadIdx.x;
    unsigned i = gid >> 6;               // node
    unsigned f = (gid & 63u) << 2;       // feature*4
    if (i >= NNODES) return;
    float di = dinv[i];
    float s  = di * di;
    const float4 v = *reinterpret_cast<const float4*>(t + (i << 8) + f);
    const float4 b = *reinterpret_cast<const float4*>(bias + f);
    float4 o;
    o.x = s * v.x + b.x; o.y = s * v.y + b.y;
    o.z = s * v.z + b.z; o.w = s * v.w + b.w;
    *reinterpret_cast<float4*>(agg + (i << 8) + f) = o;
}

// One wave per edge: uniform index/dinv loads, each lane handles 2 float4 chunks.
__global__ __launch_bounds__(256) void edge_scatter_kernel(const int* __restrict__ ei,
                                                           const float* __restrict__ t,
                                                           const float* __restrict__ dinv,
                                                           float* __restrict__ agg) {
    const int wid  = (int)((blockIdx.x * 256u + threadIdx.x) >> 5);
    const int lane = threadIdx.x & 31;
    if (wid >= NEDGES) return;
    const int src = ei[wid];
    const int dst = ei[NEDGES + wid];
    const float w = dinv[src] * dinv[dst];
    const float4* __restrict__ vp = reinterpret_cast<const float4*>(t + ((unsigned)src << 8));
    float* __restrict__ ap = agg + ((unsigned)dst << 8);
#pragma unroll
    for (int j = 0; j < 2; ++j) {
        const int f4 = lane + (j << 5);          // float4 index, 64 per row
        const float4 v = vp[f4];
        float* p = ap + (f4 << 2);
        atomicAdd(p + 0, w * v.x);
        atomicAdd(p + 1, w * v.y);
        atomicAdd(p + 2, w * v.z);
        atomicAdd(p + 3, w * v.w);
    }
}

// ---------------- layer 2 (256 -> 1) ----------------
// z[i] = relu(a[i]) . W2   (one wave per node, wave32 reduction)
__global__ __launch_bounds__(256) void dot_relu_kernel(const float* __restrict__ a,
                                                       const float* __restrict__ W2,
                                                       float* __restrict__ z) {
    int wave = (int)((blockIdx.x * 256u + threadIdx.x) >> 5);
    int lane = threadIdx.x & 31;
    if (wave >= NNODES) return;
    const float* __restrict__ row = a + ((unsigned)wave << 8);
    float s = 0.0f;
#pragma unroll
    for (int j = 0; j < FDIM / 32; ++j) {
        int f = lane + (j << 5);
        s += fmaxf(row[f], 0.0f) * W2[f];
    }
#pragma unroll
    for (int m = 16; m >= 1; m >>= 1) s += __shfl_xor(s, m, 32);
    if (lane == 0) z[wave] = s;
}

__global__ __launch_bounds__(256) void out_init_kernel(const float* __restrict__ z,
                                                       const float* __restrict__ dinv,
                                                       const float* __restrict__ b2,
                                                       float* __restrict__ out) {
    int i = blockIdx.x * blockDim.x + threadIdx.x;
    if (i < NNODES) {
        float di = dinv[i];
        out[i] = di * di * z[i] + b2[0];
    }
}

__global__ __launch_bounds__(256) void out_scatter_kernel(const int* __restrict__ ei,
                                                          const float* __restrict__ z,
                                                          const float* __restrict__ dinv,
                                                          float* __restrict__ out) {
    int e = blockIdx.x * blockDim.x + threadIdx.x;
    if (e < NEDGES) {
        int src = ei[e];
        int dst = ei[NEDGES + e];
        atomicAdd(&out[dst], dinv[src] * dinv[dst] * z[src]);
    }
}

// ---------------- host launch ----------------

extern "C" void kernel_launch(void* const* d_in, const int* in_sizes, int n_in,
                              void* d_out, int out_size, void* d_ws, size_t ws_size,
                              hipStream_t stream) {
    const float* x   = (const float*)d_in[0];
    const int*   ei  = (const int*)  d_in[1];
    const float* W0  = (const float*)d_in[2];
    const float* b0  = (const float*)d_in[3];
    const float* W1  = (const float*)d_in[4];
    const float* b1  = (const float*)d_in[5];
    const float* W2  = (const float*)d_in[6];
    const float* b2  = (const float*)d_in[7];
    float* out = (float*)d_out;

    char* ws = (char*)d_ws;
    const size_t dinv_bytes = ((size_t)NNODES * 4 + 255) & ~(size_t)255;
    float* dinv = (float*)ws;
    float* tbuf = (float*)(ws + dinv_bytes);                                  // N x 256
    float* abuf = (float*)(ws + dinv_bytes + (size_t)NNODES * FDIM * 4);      // N x 256

    const int T = 256;
    const int nodeBlocks = (NNODES + T - 1) / T;
    const int edgeBlocks = (NEDGES + T - 1) / T;
    const int nf4Blocks  = (int)(((long long)NNODES * 64 + T - 1) / T);
    const int gemmWaves  = MPAIRS * 4;                       // 6252 waves
    const int gemmBlocks = (gemmWaves + (T / 32) - 1) / (T / 32);
    const int edgeWaveBlocks = (int)(((long long)NEDGES * 32 + T - 1) / T);  // wave/edge
    const int nodeWaveBlocks = (NNODES + (T / 32) - 1) / (T / 32);

    // normalization: deg -> dinv (in place)
    deg_init_kernel<<<nodeBlocks, T, 0, stream>>>(dinv);
    deg_count_kernel<<<edgeBlocks, T, 0, stream>>>(ei, dinv);
    dinv_kernel<<<nodeBlocks, T, 0, stream>>>(dinv);

    // ---- layer 0: t = x @ W0 ; a = Ahat*t + b0 (ReLU folded into next consumer)
    gemm_wmma_f32_kernel<0><<<gemmBlocks, T, 0, stream>>>(x, W0, tbuf);
    agg_init_kernel<<<nf4Blocks, T, 0, stream>>>(tbuf, dinv, b0, abuf);
    edge_scatter_kernel<<<edgeWaveBlocks, T, 0, stream>>>(ei, tbuf, dinv, abuf);

    // ---- layer 1: t = relu(a) @ W1 ; a = Ahat*t + b1
    gemm_wmma_f32_kernel<1><<<gemmBlocks, T, 0, stream>>>(abuf, W1, tbuf);
    agg_init_kernel<<<nf4Blocks, T, 0, stream>>>(tbuf, dinv, b1, abuf);
    edge_scatter_kernel<<<edgeWaveBlocks, T, 0, stream>>>(ei, tbuf, dinv, abuf);

    // ---- layer 2: z = relu(a) @ W2 (matvec) ; out = Ahat*z + b2
    dot_relu_kernel<<<nodeWaveBlocks, T, 0, stream>>>(abuf, W2, tbuf);  // reuse tbuf
    out_init_kernel<<<nodeBlocks, T, 0, stream>>>(tbuf, dinv, b2, out);
    out_scatter_kernel<<<edgeBlocks, T, 0, stream>>>(ei, tbuf, dinv, out);
}